// GAT_54477365182589
// MI455X (gfx1250) — compile-verified
//
#include <hip/hip_runtime.h>

// GAT (2-layer, PyG-style with self-loops) for MI455X / gfx1250, wave32.
// Layer-1 node transform uses V_WMMA_F32_16X16X4_F32 (fp32-exact WMMA).
// Edge softmax uses u32-encoded float atomicMax + L2-resident f32 atomicAdd.

#define NN     50000
#define EE     800000
#define FIN    128
#define HID    32
#define HEADS  4
#define SLOPE  0.2f

typedef float v2f __attribute__((ext_vector_type(2)));
typedef float v8f __attribute__((ext_vector_type(8)));

__device__ __forceinline__ float lrelu(float x) { return x > 0.f ? x : SLOPE * x; }

// Monotonic order-preserving float<->uint encoding for atomicMax on floats.
__device__ __forceinline__ unsigned enc_f(float f) {
  unsigned u = __float_as_uint(f);
  return (u & 0x80000000u) ? ~u : (u | 0x80000000u);
}
__device__ __forceinline__ float dec_f(unsigned u) {
  return __uint_as_float((u & 0x80000000u) ? (u ^ 0x80000000u) : ~u);
}

// ---------------------------------------------------------------------------
// Layer-1 GEMM: H[N,128] = X[N,128] @ W[128,128] via V_WMMA_F32_16X16X4_F32.
// Block = 256 thr = 8 waves; block computes 16 rows x 128 cols; wave w owns
// the 16-col tile [16w,16w+16). A tile staged transposed in LDS (stride 20),
// W staged in 64-row chunks (stride 144: half-wave bank groups disjoint).
// ---------------------------------------------------------------------------
#define XS 20
#define WS 144

__global__ __launch_bounds__(256) void gat_gemm1_wmma(
    const float* __restrict__ X, const float* __restrict__ W,
    float* __restrict__ H)
{
  __shared__ float Xs[FIN * XS];   // Xs[k*XS + m] : transposed 16x128 X tile
  __shared__ float Ws[64 * WS];    // Ws[kl*WS + n]: 64x128 chunk of W

  const int tid  = threadIdx.x;
  const int row0 = blockIdx.x * 16;    // N = 50000 = 3125*16, no remainder

  for (int i = tid; i < 16 * FIN; i += 256) {
    int m = i >> 7, k = i & 127;
    Xs[k * XS + m] = X[(row0 + m) * FIN + k];
  }

  const int lane = tid & 31;
  const int wave = tid >> 5;
  const int half = lane >> 4;      // ISA: lanes 16-31 hold K+2 (A), rows M+8 (C/D)
  const int m    = lane & 15;
  const int ncol = wave * 16 + m;

  v8f acc = {};
  for (int kc = 0; kc < 2; ++kc) {
    __syncthreads();
    for (int i = tid; i < 64 * FIN; i += 256) {
      int kl = i >> 7, c = i & 127;
      Ws[kl * WS + c] = W[(kc * 64 + kl) * FIN + c];
    }
    __syncthreads();
    #pragma unroll
    for (int k0 = 0; k0 < 64; k0 += 4) {
      const int kg = kc * 64 + k0 + 2 * half;   // global K for A fragment
      const int kl = k0 + 2 * half;             // chunk-local K for B fragment
      v2f a, b;
      a.x = Xs[kg * XS + m];            // A: VGPR0 = K, VGPR1 = K+1 (per lane-half)
      a.y = Xs[(kg + 1) * XS + m];
      b.x = Ws[kl * WS + ncol];         // B: row K striped across lanes (N = lane)
      b.y = Ws[(kl + 1) * WS + ncol];
      acc = __builtin_amdgcn_wmma_f32_16x16x4_f32(false, a, false, b,
                                                  (short)0, acc, false, false);
    }
  }
  #pragma unroll
  for (int i = 0; i < 8; ++i) {        // C/D: VGPR i -> rows i (lo half) / i+8 (hi)
    H[(row0 + i + 8 * half) * FIN + ncol] = acc[i];
  }
}

// ---------------------------------------------------------------------------
// Per-node attention logits + seed segment-max with the self-loop edge.
// ---------------------------------------------------------------------------
__global__ void gat_alpha1(const float* __restrict__ H,
                           const float* __restrict__ a_src,   // [4][32]
                           const float* __restrict__ a_dst,
                           float* __restrict__ asrc, float* __restrict__ adst,
                           unsigned* __restrict__ emax)
{
  int n = blockIdx.x * blockDim.x + threadIdx.x;
  if (n >= NN) return;
  for (int h = 0; h < HEADS; ++h) {
    float s = 0.f, d = 0.f;
    const float* hp = H + n * FIN + h * HID;
    #pragma unroll 8
    for (int c = 0; c < HID; ++c) {
      float v = hp[c];
      s += v * a_src[h * HID + c];
      d += v * a_dst[h * HID + c];
    }
    asrc[n * HEADS + h] = s;
    adst[n * HEADS + h] = d;
    emax[n * HEADS + h] = enc_f(lrelu(s + d));   // self loop src==dst==n
  }
}

__global__ void gat_edge_max1(const int* __restrict__ ei,
                              const float* __restrict__ asrc,
                              const float* __restrict__ adst,
                              unsigned* __restrict__ emax)
{
  int e = blockIdx.x * blockDim.x + threadIdx.x;
  if (e >= EE) return;
  int s = ei[e], d = ei[EE + e];
  for (int h = 0; h < HEADS; ++h) {
    float ev = lrelu(asrc[s * HEADS + h] + adst[d * HEADS + h]);
    atomicMax(&emax[d * HEADS + h], enc_f(ev));
  }
}

// Init denom/acc with the self-loop term (runs after segment-max is final).
__global__ void gat_selfw1(const float* __restrict__ H,
                           const float* __restrict__ asrc,
                           const float* __restrict__ adst,
                           const unsigned* __restrict__ emax,
                           float* __restrict__ denom, float* __restrict__ acc)
{
  int idx = blockIdx.x * blockDim.x + threadIdx.x;   // n*128 + j
  if (idx >= NN * FIN) return;
  int n = idx >> 7, j = idx & 127;
  int h = j >> 5;
  float es = lrelu(asrc[n * HEADS + h] + adst[n * HEADS + h]);
  float w  = expf(es - dec_f(emax[n * HEADS + h]));
  acc[idx] = H[idx] * w;
  if ((j & 31) == 0) denom[n * HEADS + h] = w;
}

// One wave32 per edge: 32 channels per head = one coalesced atomic burst.
__global__ __launch_bounds__(256) void gat_edge_acc1(
    const int* __restrict__ ei, const float* __restrict__ H,
    const float* __restrict__ asrc, const float* __restrict__ adst,
    const unsigned* __restrict__ emax,
    float* __restrict__ denom, float* __restrict__ acc)
{
  int e = blockIdx.x * 8 + (threadIdx.x >> 5);
  if (e >= EE) return;
  int lane = threadIdx.x & 31;
  int s = ei[e], d = ei[EE + e];
  for (int h = 0; h < HEADS; ++h) {
    float ev = lrelu(asrc[s * HEADS + h] + adst[d * HEADS + h]);
    float w  = expf(ev - dec_f(emax[d * HEADS + h]));
    if (lane == 0) atomicAdd(&denom[d * HEADS + h], w);
    int c = h * HID + lane;
    atomicAdd(&acc[d * FIN + c], H[s * FIN + c] * w);
  }
}

__global__ void gat_finish1(const float* __restrict__ acc,
                            const float* __restrict__ denom,
                            const float* __restrict__ b1,
                            float* __restrict__ X2)
{
  int idx = blockIdx.x * blockDim.x + threadIdx.x;   // n*32 + c
  if (idx >= NN * HID) return;
  int n = idx >> 5, c = idx & 31;
  float s = 0.f;
  #pragma unroll
  for (int h = 0; h < HEADS; ++h)
    s += acc[n * FIN + h * HID + c] / denom[n * HEADS + h];
  float v = 0.25f * s + b1[c];                       // mean over heads + bias
  X2[idx] = v > 0.f ? v : expm1f(v);                 // ELU
}

// ---------------------------------------------------------------------------
// Layer 2 (1 head, 1 channel)
// ---------------------------------------------------------------------------
__global__ void gat_l2node(const float* __restrict__ X2,
                           const float* __restrict__ W2,
                           const float* __restrict__ as2,
                           const float* __restrict__ ad2,
                           float* __restrict__ H2, float* __restrict__ asrc,
                           float* __restrict__ adst, unsigned* __restrict__ emax)
{
  int n = blockIdx.x * blockDim.x + threadIdx.x;
  if (n >= NN) return;
  float s = 0.f;
  #pragma unroll 8
  for (int c = 0; c < HID; ++c) s += X2[n * HID + c] * W2[c];
  H2[n] = s;
  float a = s * as2[0], b = s * ad2[0];
  asrc[n] = a; adst[n] = b;
  emax[n] = enc_f(lrelu(a + b));
}

__global__ void gat_edge_max2(const int* __restrict__ ei,
                              const float* __restrict__ asrc,
                              const float* __restrict__ adst,
                              unsigned* __restrict__ emax)
{
  int e = blockIdx.x * blockDim.x + threadIdx.x;
  if (e >= EE) return;
  atomicMax(&emax[ei[EE + e]], enc_f(lrelu(asrc[ei[e]] + adst[ei[EE + e]])));
}

__global__ void gat_selfw2(const float* __restrict__ H2,
                           const float* __restrict__ asrc,
                           const float* __restrict__ adst,
                           const unsigned* __restrict__ emax,
                           float* __restrict__ denom, float* __restrict__ acc)
{
  int n = blockIdx.x * blockDim.x + threadIdx.x;
  if (n >= NN) return;
  float w = expf(lrelu(asrc[n] + adst[n]) - dec_f(emax[n]));
  denom[n] = w;
  acc[n]   = H2[n] * w;
}

__global__ void gat_edge_acc2(const int* __restrict__ ei,
                              const float* __restrict__ H2,
                              const float* __restrict__ asrc,
                              const float* __restrict__ adst,
                              const unsigned* __restrict__ emax,
                              float* __restrict__ denom, float* __restrict__ acc)
{
  int e = blockIdx.x * blockDim.x + threadIdx.x;
  if (e >= EE) return;
  int s = ei[e], d = ei[EE + e];
  float w = expf(lrelu(asrc[s] + adst[d]) - dec_f(emax[d]));
  atomicAdd(&denom[d], w);
  atomicAdd(&acc[d], H2[s] * w);
}

__global__ void gat_finish2(const float* __restrict__ acc,
                            const float* __restrict__ denom,
                            const float* __restrict__ b2,
                            float* __restrict__ out)
{
  int n = blockIdx.x * blockDim.x + threadIdx.x;
  if (n >= NN) return;
  out[n] = acc[n] / denom[n] + b2[0];
}

// ---------------------------------------------------------------------------
extern "C" void kernel_launch(void* const* d_in, const int* in_sizes, int n_in,
                              void* d_out, int out_size, void* d_ws, size_t ws_size,
                              hipStream_t stream)
{
  const float* x    = (const float*)d_in[0];
  const int*   ei   = (const int*)d_in[1];   // edge_index [2,E] (int32 on device)
  const float* W1   = (const float*)d_in[3];
  const float* asc1 = (const float*)d_in[4];
  const float* adc1 = (const float*)d_in[5];
  const float* b1   = (const float*)d_in[6];
  const float* W2   = (const float*)d_in[7];
  const float* asc2 = (const float*)d_in[8];
  const float* adc2 = (const float*)d_in[9];
  const float* b2   = (const float*)d_in[10];
  float* out = (float*)d_out;

  // workspace carve-up (256B aligned)
  char* p = (char*)d_ws;
  auto take = [&](size_t bytes) {
    char* r = p;
    p += (bytes + 255) & ~size_t(255);
    return r;
  };
  float*    h1     = (float*)   take(sizeof(float) * NN * FIN);
  float*    as1    = (float*)   take(sizeof(float) * NN * HEADS);
  float*    ad1    = (float*)   take(sizeof(float) * NN * HEADS);
  unsigned* emax1  = (unsigned*)take(sizeof(unsigned) * NN * HEADS);
  float*    den1   = (float*)   take(sizeof(float) * NN * HEADS);
  float*    acc1   = (float*)   take(sizeof(float) * NN * FIN);
  float*    x2     = (float*)   take(sizeof(float) * NN * HID);
  float*    h2     = (float*)   take(sizeof(float) * NN);
  float*    as2    = (float*)   take(sizeof(float) * NN);
  float*    ad2    = (float*)   take(sizeof(float) * NN);
  unsigned* emax2  = (unsigned*)take(sizeof(unsigned) * NN);
  float*    den2   = (float*)   take(sizeof(float) * NN);
  float*    acc2   = (float*)   take(sizeof(float) * NN);

  const int nodeBlocks = (NN + 255) / 256;          // 196
  const int edgeBlocks = (EE + 255) / 256;          // 3125

  // ---- layer 1 ----
  gat_gemm1_wmma<<<NN / 16, 256, 0, stream>>>(x, W1, h1);                       // 3125 blocks
  gat_alpha1<<<nodeBlocks, 256, 0, stream>>>(h1, asc1, adc1, as1, ad1, emax1);
  gat_edge_max1<<<edgeBlocks, 256, 0, stream>>>(ei, as1, ad1, emax1);
  gat_selfw1<<<(NN * FIN + 255) / 256, 256, 0, stream>>>(h1, as1, ad1, emax1, den1, acc1);
  gat_edge_acc1<<<(EE + 7) / 8, 256, 0, stream>>>(ei, h1, as1, ad1, emax1, den1, acc1);
  gat_finish1<<<(NN * HID + 255) / 256, 256, 0, stream>>>(acc1, den1, b1, x2);

  // ---- layer 2 ----
  gat_l2node<<<nodeBlocks, 256, 0, stream>>>(x2, W2, asc2, adc2, h2, as2, ad2, emax2);
  gat_edge_max2<<<edgeBlocks, 256, 0, stream>>>(ei, as2, ad2, emax2);
  gat_selfw2<<<nodeBlocks, 256, 0, stream>>>(h2, as2, ad2, emax2, den2, acc2);
  gat_edge_acc2<<<edgeBlocks, 256, 0, stream>>>(ei, h2, as2, ad2, emax2, den2, acc2);
  gat_finish2<<<nodeBlocks, 256, 0, stream>>>(acc2, den2, b2, out);

  (void)in_sizes; (void)n_in; (void)out_size; (void)ws_size;
}